// CoherenceNetWithGCN_49778670961019
// MI455X (gfx1250) — compile-verified
//
#include <hip/hip_runtime.h>
#include <hip/hip_bf16.h>

#define H   128
#define ATT 128
#define NS  1024
#define NE  512
#define M_  1024

typedef __attribute__((ext_vector_type(2))) float v2f;
typedef __attribute__((ext_vector_type(8))) float v8f;

__device__ __forceinline__ float fast_tanh(float x) {
    // tanh(x) = (e^{2x}-1)/(e^{2x}+1), via v_exp_f32 (exp2). Clamp to avoid inf/inf.
    float xc = fminf(fmaxf(x, -12.0f), 12.0f);
    float e2 = __builtin_amdgcn_exp2f(xc * 2.8853900817779268f); // 2*log2(e)
    return (e2 - 1.0f) / (e2 + 1.0f);
}

__device__ __forceinline__ float fast_exp(float x) {
    return __builtin_amdgcn_exp2f(x * 1.4426950408889634f);
}

// ---------------------------------------------------------------------------
// D[n, o] = sum_i X[n,i] * W[o, coloff + i]  (+ bias[o])      (one wave / tile)
// X: [*, H] row-major, W row stride ldw, D: [*, H] row-major.
// grid.x = (Nrows/16) * (H/16)
// ---------------------------------------------------------------------------
__global__ __launch_bounds__(32) void wmma_proj(const float* __restrict__ X,
                                                const float* __restrict__ W,
                                                const float* __restrict__ bias,
                                                float* __restrict__ D,
                                                int ldw, int coloff) {
    const int tilesO = H / 16;
    const int n0 = (blockIdx.x / tilesO) * 16;
    const int o0 = (blockIdx.x % tilesO) * 16;
    const int lane = threadIdx.x;
    const int half = lane >> 4, l16 = lane & 15;

    const float* xrow = X + (size_t)(n0 + l16) * H;
    const float* wrow = W + (size_t)(o0 + l16) * ldw + coloff;

    v8f c = {};
    for (int k = 0; k < H; k += 4) {
        const int kb = k + 2 * half;                 // lanes 16-31 hold K+2,K+3
        v2f a = *(const v2f*)(xrow + kb);            // A[M=n0+l16, K=kb..kb+1]
        v2f b = *(const v2f*)(wrow + kb);            // B[K, N=o0+l16] = W[o, k]
        c = __builtin_amdgcn_wmma_f32_16x16x4_f32(false, a, false, b,
                                                  (short)0, c, false, false);
    }
#pragma unroll
    for (int v = 0; v < 8; ++v) {
        const int row = n0 + v + 8 * half;           // C layout: lanes16-31 -> M+8
        const int col = o0 + l16;
        float out = c[v] + (bias ? bias[col] : 0.0f);
        D[(size_t)row * H + col] = out;
    }
}

// ---------------------------------------------------------------------------
// Per attender column m: S[n] = bs + sum_h ws[h]*tanh(A[n,h] + B[m,h]);
// softmax over n; write Wt[m, n] (transposed -> contiguous stores & WMMA loads)
// grid.x = M_, block = 256 (8 waves)
// ---------------------------------------------------------------------------
__global__ __launch_bounds__(256) void score_softmax(const float* __restrict__ A,
                                                     const float* __restrict__ Bv,
                                                     const float* __restrict__ ws,
                                                     const float* __restrict__ bs,
                                                     float* __restrict__ Wt,
                                                     int N) {
    __shared__ float sB[H];
    __shared__ float sW[H];
    __shared__ float sS[NS];
    __shared__ float red[8];

    const int m = blockIdx.x;
    const int t = threadIdx.x;
    const int wave = t >> 5, lane = t & 31;

    if (t < H) { sB[t] = Bv[(size_t)m * H + t]; sW[t] = ws[t]; }
    __syncthreads();

    const float bias = bs[0];
    // one wave per attendee row: coalesced A-row reads, shfl reduce over h
    for (int n = wave; n < N; n += 8) {
        const float* arow = A + (size_t)n * H;
        float p = 0.0f;
        for (int h = lane; h < H; h += 32)
            p += sW[h] * fast_tanh(arow[h] + sB[h]);
#pragma unroll
        for (int off = 16; off; off >>= 1) p += __shfl_xor(p, off, 32);
        if (lane == 0) sS[n] = p + bias;
    }
    __syncthreads();

    // block max over sS[0..N)
    float mx = -INFINITY;
    for (int n = t; n < N; n += 256) mx = fmaxf(mx, sS[n]);
#pragma unroll
    for (int off = 16; off; off >>= 1) mx = fmaxf(mx, __shfl_xor(mx, off, 32));
    if (lane == 0) red[wave] = mx;
    __syncthreads();
    if (t == 0) {
        float v = red[0];
        for (int i = 1; i < 8; ++i) v = fmaxf(v, red[i]);
        red[0] = v;
    }
    __syncthreads();
    mx = red[0];
    __syncthreads();

    // exp + block sum
    float ps = 0.0f;
    for (int n = t; n < N; n += 256) {
        float e = fast_exp(sS[n] - mx);
        sS[n] = e;
        ps += e;
    }
#pragma unroll
    for (int off = 16; off; off >>= 1) ps += __shfl_xor(ps, off, 32);
    if (lane == 0) red[wave] = ps;
    __syncthreads();
    if (t == 0) {
        float v = 0.0f;
        for (int i = 0; i < 8; ++i) v += red[i];
        red[0] = v;
    }
    __syncthreads();
    const float inv = 1.0f / red[0];

    for (int n = t; n < N; n += 256)
        Wt[(size_t)m * N + n] = sS[n] * inv;     // contiguous per block
}

// ---------------------------------------------------------------------------
// ctx[m, h] = sum_n Wt[m, n] * X[n, h]          (one wave / 16x16 tile)
// grid.x = (M_/16) * (H/16)
// ---------------------------------------------------------------------------
__global__ __launch_bounds__(32) void wmma_ctx(const float* __restrict__ Wt,
                                               const float* __restrict__ X,
                                               float* __restrict__ Cout,
                                               int Ncols) {
    const int tilesH = H / 16;
    const int m0 = (blockIdx.x / tilesH) * 16;
    const int h0 = (blockIdx.x % tilesH) * 16;
    const int lane = threadIdx.x;
    const int half = lane >> 4, l16 = lane & 15;

    const float* wrow = Wt + (size_t)(m0 + l16) * Ncols;

    v8f c = {};
    for (int k = 0; k < Ncols; k += 4) {
        const int kb = k + 2 * half;
        v2f a = *(const v2f*)(wrow + kb);            // A[m, k] contiguous
        v2f b;
        b.x = X[(size_t)kb * H + h0 + l16];          // B[k, h] = X[k, h]
        b.y = X[(size_t)(kb + 1) * H + h0 + l16];
        c = __builtin_amdgcn_wmma_f32_16x16x4_f32(false, a, false, b,
                                                  (short)0, c, false, false);
    }
#pragma unroll
    for (int v = 0; v < 8; ++v)
        Cout[(size_t)(m0 + v + 8 * half) * H + h0 + l16] = c[v];
}

// ---------------------------------------------------------------------------
// att[m, a] = tanh( sum_k feats[m,k]*W_lin[a,k] + b_lin[a] ),
// feats = [attender | ctx_s | ctx_e], K = 3H (segments never straddled, k%4==0)
// grid.x = (M_/16) * (ATT/16)
// ---------------------------------------------------------------------------
__global__ __launch_bounds__(32) void wmma_final(const float* __restrict__ attender,
                                                 const float* __restrict__ ctx_s,
                                                 const float* __restrict__ ctx_e,
                                                 const float* __restrict__ W_lin,
                                                 const float* __restrict__ b_lin,
                                                 float* __restrict__ att) {
    const int tilesA = ATT / 16;
    const int m0 = (blockIdx.x / tilesA) * 16;
    const int a0 = (blockIdx.x % tilesA) * 16;
    const int lane = threadIdx.x;
    const int half = lane >> 4, l16 = lane & 15;

    const float* wrow = W_lin + (size_t)(a0 + l16) * (3 * H);

    v8f c = {};
    for (int k = 0; k < 3 * H; k += 4) {
        const int kb = k + 2 * half;
        const float* seg; int kk;
        if (k < H)            { seg = attender; kk = kb; }
        else if (k < 2 * H)   { seg = ctx_s;    kk = kb - H; }
        else                  { seg = ctx_e;    kk = kb - 2 * H; }
        v2f a = *(const v2f*)(seg + (size_t)(m0 + l16) * H + kk);
        v2f b = *(const v2f*)(wrow + kb);
        c = __builtin_amdgcn_wmma_f32_16x16x4_f32(false, a, false, b,
                                                  (short)0, c, false, false);
    }
#pragma unroll
    for (int v = 0; v < 8; ++v) {
        const int row = m0 + v + 8 * half;
        const int col = a0 + l16;
        att[(size_t)row * ATT + col] = fast_tanh(c[v] + b_lin[col]);
    }
}

// ---------------------------------------------------------------------------
// out[m] = sum_a att[m,a]*W_coh[a] + b_coh     (one wave per row)
// grid.x = M_/8, block = 256
// ---------------------------------------------------------------------------
__global__ __launch_bounds__(256) void coherence_kernel(const float* __restrict__ att,
                                                        const float* __restrict__ Wcoh,
                                                        const float* __restrict__ bcoh,
                                                        float* __restrict__ out) {
    const int wave = threadIdx.x >> 5, lane = threadIdx.x & 31;
    const int m = blockIdx.x * 8 + wave;
    float p = 0.0f;
    for (int a = lane; a < ATT; a += 32)
        p += att[(size_t)m * ATT + a] * Wcoh[a];
#pragma unroll
    for (int off = 16; off; off >>= 1) p += __shfl_xor(p, off, 32);
    if (lane == 0) out[m] = p + bcoh[0];
}

extern "C" void kernel_launch(void* const* d_in, const int* in_sizes, int n_in,
                              void* d_out, int out_size, void* d_ws, size_t ws_size,
                              hipStream_t stream) {
    (void)in_sizes; (void)n_in; (void)out_size; (void)ws_size;

    const float* stmts    = (const float*)d_in[0];   // [NS, H]
    const float* eres     = (const float*)d_in[1];   // [NE, H]
    const float* attender = (const float*)d_in[2];   // [M, H]
    const float* Wc_s     = (const float*)d_in[3];   // [H, 2H]
    const float* bc_s     = (const float*)d_in[4];
    const float* ws_s     = (const float*)d_in[5];
    const float* bs_s     = (const float*)d_in[6];
    const float* Wc_e     = (const float*)d_in[7];   // [H, 2H]
    const float* bc_e     = (const float*)d_in[8];
    const float* ws_e     = (const float*)d_in[9];
    const float* bs_e     = (const float*)d_in[10];
    const float* W_lin    = (const float*)d_in[11];  // [ATT, 3H]
    const float* b_lin    = (const float*)d_in[12];
    const float* W_coh    = (const float*)d_in[13];  // [1, ATT]
    const float* b_coh    = (const float*)d_in[14];
    float* out = (float*)d_out;                      // [M, 1]

    // workspace layout (floats), total ~9.3 MB
    float* ws = (float*)d_ws;
    float* A_s   = ws;                  // NS*H
    float* B_s   = A_s  + NS * H;       // M*H
    float* A_e   = B_s  + M_ * H;       // NE*H
    float* B_e   = A_e  + NE * H;       // M*H
    float* Wt_ss = B_e  + M_ * H;       // M*NS
    float* Wt_es = Wt_ss + (size_t)M_ * NS;  // M*NE
    float* ctx_s = Wt_es + (size_t)M_ * NE;  // M*H
    float* ctx_e = ctx_s + M_ * H;      // M*H
    float* attv  = ctx_e + M_ * H;      // M*ATT

    // 1) projections (f32 WMMA)
    wmma_proj<<<(NS / 16) * (H / 16), 32, 0, stream>>>(stmts,    Wc_s, bc_s,   A_s, 2 * H, 0);
    wmma_proj<<<(M_ / 16) * (H / 16), 32, 0, stream>>>(attender, Wc_s, nullptr, B_s, 2 * H, H);
    wmma_proj<<<(NE / 16) * (H / 16), 32, 0, stream>>>(eres,     Wc_e, bc_e,   A_e, 2 * H, 0);
    wmma_proj<<<(M_ / 16) * (H / 16), 32, 0, stream>>>(attender, Wc_e, nullptr, B_e, 2 * H, H);

    // 2) fused tanh-score + softmax over attendees (VALU/TRANS bound)
    score_softmax<<<M_, 256, 0, stream>>>(A_s, B_s, ws_s, bs_s, Wt_ss, NS);
    score_softmax<<<M_, 256, 0, stream>>>(A_e, B_e, ws_e, bs_e, Wt_es, NE);

    // 3) context GEMMs (f32 WMMA, K = NS / NE)
    wmma_ctx<<<(M_ / 16) * (H / 16), 32, 0, stream>>>(Wt_ss, stmts, ctx_s, NS);
    wmma_ctx<<<(M_ / 16) * (H / 16), 32, 0, stream>>>(Wt_es, eres,  ctx_e, NE);

    // 4) combine + tanh (f32 WMMA over segmented concat, K = 3H)
    wmma_final<<<(M_ / 16) * (ATT / 16), 32, 0, stream>>>(attender, ctx_s, ctx_e,
                                                          W_lin, b_lin, attv);

    // 5) coherence head
    coherence_kernel<<<M_ / 8, 256, 0, stream>>>(attv, W_coh, b_coh, out);
}